// ECGate_70875550319381
// MI455X (gfx1250) — compile-verified
//
#include <hip/hip_runtime.h>
#include <hip/hip_bf16.h>

// ECGate for MI455X (gfx1250, wave32, WMMA).
// Phases:
//  K1: fused gating GEMM (fp32 WMMA 16x16x4) + softmax -> affinity[b*8+e][4096] (ws)
//  K2: per-row bitonic sort (value desc, index asc tiebreak) in LDS -> top-512 sorted
//  K3: 512MB one-hot dispatch mask via non-temporal b128 stores (streaming, bypass cache)

#define BSZ_ 8
#define SEQ_ 4096
#define HID_ 1024
#define NE_ 8
#define CAP_ 512
#define SCALE_ 1.0f

typedef __attribute__((ext_vector_type(2))) float v2f;
typedef __attribute__((ext_vector_type(4))) float v4f;
typedef __attribute__((ext_vector_type(8))) float v8f;

// ---------------- Kernel 1: GEMM (WMMA f32 16x16x4) + softmax -------------
// Grid: 256 blocks x 256 threads. Each block: 128 tokens, 8 waves x 16 rows.
__global__ __launch_bounds__(256) void ecgate_gemm_softmax(
    const float* __restrict__ hs,   // [BSZ*SEQ, HID]
    const float* __restrict__ w,    // [HID, NE]
    float* __restrict__ aff)        // [BSZ*NE, SEQ]
{
    __shared__ float w_lds[HID_ * NE_];   // 32 KB, [k][e]
    __shared__ float a_lds[128 * 36];     // 18 KB, padded stride 36 (bank-conflict-free)
    __shared__ float l_lds[128 * NE_];    // 4 KB logits staging

    const int tid  = threadIdx.x;
    const int wave = tid >> 5;
    const int lane = tid & 31;
    const int hi   = lane >> 4;   // half-wave select
    const int l15  = lane & 15;
    const int l7   = lane & 7;

    // Stage the whole gating weight panel once (8192 floats).
    for (int i = tid; i < HID_ * NE_; i += 256) w_lds[i] = w[i];

    const long tok0 = (long)blockIdx.x * 128;
    const float* hbase = hs + tok0 * HID_;

    v8f c = {};  // 16x16 fp32 accumulator (cols 8..15 are don't-care)

    for (int kb = 0; kb < HID_; kb += 32) {
        __syncthreads();
        // Cooperative coalesced load of A tile: 128 rows x 32 floats (b128 each).
        #pragma unroll
        for (int i = 0; i < 4; ++i) {
            int idx = tid + 256 * i;          // float4 slot 0..1023
            int row = idx >> 3;
            int c4  = idx & 7;
            const float* gp = hbase + (long)row * HID_ + kb + c4 * 4;
            if (kb + 32 < HID_) __builtin_prefetch(gp + 32, 0, 1);
            v4f v = *(const v4f*)gp;
            *(v4f*)(a_lds + row * 36 + c4 * 4) = v;
        }
        __syncthreads();

        const float* arow = a_lds + (wave * 16 + l15) * 36 + 2 * hi;
        #pragma unroll
        for (int kk = 0; kk < 32; kk += 4) {
            // A frag (16x4 f32): lane row M=l15; vgpr0 = K=kk+2*hi, vgpr1 = K+1
            v2f a = *(const v2f*)(arow + kk);
            // B frag (4x16 f32): vgpr0 lanes0-15 K=kk, lanes16-31 K=kk+2; vgpr1 = K+1
            int kA = kb + kk + 2 * hi;
            v2f b;
            b.x = w_lds[kA * NE_ + l7];         // cols 8..15 mirror 0..7 (ignored)
            b.y = w_lds[(kA + 1) * NE_ + l7];
            c = __builtin_amdgcn_wmma_f32_16x16x4_f32(
                    false, a, false, b, (short)0, c, false, false);
        }
    }

    // Spill logits (cols 0..7 only) to LDS: token-major [128][8].
    if (l15 < 8) {
        int tl0 = wave * 16 + 8 * hi;   // VGPR i -> row tl0 + i
        #pragma unroll
        for (int i = 0; i < 8; ++i)
            l_lds[(tl0 + i) * NE_ + l15] = c[i];
    }
    __syncthreads();

    // Softmax over the 8 experts; write transposed affinity [b][e][s].
    if (tid < 128) {
        float v[NE_];
        float m = -3.402823466e+38f;
        #pragma unroll
        for (int e = 0; e < NE_; ++e) { v[e] = l_lds[tid * NE_ + e]; m = fmaxf(m, v[e]); }
        float s = 0.f;
        #pragma unroll
        for (int e = 0; e < NE_; ++e) { v[e] = __expf(v[e] - m); s += v[e]; }
        float inv = 1.f / s;
        long g = tok0 + tid;
        int  b = (int)(g >> 12);
        int  sp = (int)(g & 4095);
        float* ab = aff + ((long)b * NE_) * SEQ_ + sp;
        #pragma unroll
        for (int e = 0; e < NE_; ++e) ab[(long)e * SEQ_] = v[e] * inv;
    }
}

// ---------------- Kernel 2: bitonic top-512 (sorted) per (b,e) ------------
// Grid: 64 blocks x 512 threads. Full 4096-element key-value bitonic sort in LDS.
__global__ __launch_bounds__(512) void ecgate_topk(
    const float* __restrict__ aff,
    float* __restrict__ out_idx,    // [64,512] indices as float
    float* __restrict__ out_scr,    // [64,512] scores
    int*   __restrict__ idx_ws)     // [64,512] indices as int (for K3)
{
    __shared__ float sv[SEQ_];
    __shared__ int   si[SEQ_];
    const int row = blockIdx.x;
    const int tid = threadIdx.x;
    const float* a = aff + (long)row * SEQ_;

    for (int i = tid; i < SEQ_; i += 512) { sv[i] = a[i]; si[i] = i; }
    __syncthreads();

    for (int k = 2; k <= SEQ_; k <<= 1) {
        for (int j = k >> 1; j > 0; j >>= 1) {
            for (int i = tid; i < SEQ_; i += 512) {
                int ixj = i ^ j;
                if (ixj > i) {
                    bool desc = ((i & k) == 0);  // region wants descending order
                    float va = sv[i], vb = sv[ixj];
                    int   ia = si[i], ib = si[ixj];
                    bool agtb = (va > vb) || (va == vb && ia < ib);
                    if (desc != agtb) {
                        sv[i] = vb; sv[ixj] = va;
                        si[i] = ib; si[ixj] = ia;
                    }
                }
            }
            __syncthreads();
        }
    }
    // k==SEQ final merge used desc for all i -> sv[0..] globally descending.
    if (tid < CAP_) {
        out_idx[row * CAP_ + tid] = (float)si[tid];
        out_scr[row * CAP_ + tid] = sv[tid] * SCALE_;
        idx_ws[row * CAP_ + tid]  = si[tid];
    }
}

// ---------------- Kernel 3: one-hot dispatch mask (512 MB, NT stores) -----
// Grid: 32768 blocks (one per (b,e,cap) row) x 256 threads; 4x b128 NT per thread.
__global__ __launch_bounds__(256) void ecgate_mask(
    const int* __restrict__ idx_ws,
    float* __restrict__ mask)       // [64*512, 4096]
{
    const int rc  = blockIdx.x;
    const int hot = idx_ws[rc];     // scalar-broadcast load
    float* base = mask + (long)rc * SEQ_;
    const int tid = threadIdx.x;
    #pragma unroll
    for (int i = 0; i < 4; ++i) {
        int p = (tid + 256 * i) * 4;
        v4f v;
        v.x = (p + 0 == hot) ? 1.0f : 0.0f;
        v.y = (p + 1 == hot) ? 1.0f : 0.0f;
        v.z = (p + 2 == hot) ? 1.0f : 0.0f;
        v.w = (p + 3 == hot) ? 1.0f : 0.0f;
        __builtin_nontemporal_store(v, (v4f*)(base + p));
    }
}

extern "C" void kernel_launch(void* const* d_in, const int* in_sizes, int n_in,
                              void* d_out, int out_size, void* d_ws, size_t ws_size,
                              hipStream_t stream) {
    const float* hs = (const float*)d_in[0];   // [8,4096,1024] f32
    const float* w  = (const float*)d_in[1];   // [1024,8] f32

    float* out      = (float*)d_out;
    float* out_idx  = out;                                   // [8,8,512]
    float* out_scr  = out + (long)BSZ_ * NE_ * CAP_;         // [8,8,512]
    float* out_mask = out + 2L * BSZ_ * NE_ * CAP_;          // [8,8,512,4096]

    float* aff    = (float*)d_ws;                            // 64*4096 f32 = 1 MB
    int*   idx_ws = (int*)((char*)d_ws + (size_t)BSZ_ * NE_ * SEQ_ * sizeof(float));

    ecgate_gemm_softmax<<<(BSZ_ * SEQ_) / 128, 256, 0, stream>>>(hs, w, aff);
    ecgate_topk<<<BSZ_ * NE_, 512, 0, stream>>>(aff, out_idx, out_scr, idx_ws);
    ecgate_mask<<<BSZ_ * NE_ * CAP_, 256, 0, stream>>>(idx_ws, out_mask);
}